// GDN_48576080118035
// MI455X (gfx1250) — compile-verified
//
#include <hip/hip_runtime.h>
#include <math.h>

// ---------------------------------------------------------------------------
// GDN-style graph attention pipeline for MI455X (gfx1250, wave32).
// Main GEMM z = data @ fc_w^T runs on V_WMMA_F32_16X16X4_F32 (exact f32;
// workload is HBM-bound at ~100MB / 23.3TB/s so no reason to drop precision).
// fc_w -> LDS staging uses GLOBAL_LOAD_ASYNC_TO_LDS_B64 (ASYNCcnt path);
// builtin signature (per hipcc diagnostic): (v2i __device__*, v2i __shared__*,
// imm offset, imm cpol).
// ---------------------------------------------------------------------------

typedef float v2f __attribute__((ext_vector_type(2)));
typedef float v8f __attribute__((ext_vector_type(8)));
typedef int   v2i __attribute__((ext_vector_type(2)));
typedef __attribute__((address_space(1))) v2i* gv2i_p;  // global (device) ptr
typedef __attribute__((address_space(3))) v2i* lv2i_p;  // LDS ptr

#define Bd   128
#define Nd   512
#define Dd   64
#define TOPK 20
#define PAIRS 65536            // B*N
#define AGG_WAVES 2048         // 256 blocks * 8 waves
#define PAIRS_PER_WAVE 32

// ---------------------------------------------------------------- norms ----
__global__ void norms_kernel(const float* __restrict__ emb,
                             float* __restrict__ nrm) {
  int i = blockIdx.x * blockDim.x + threadIdx.x;
  if (i < Nd) {
    float s = 0.f;
    #pragma unroll
    for (int d = 0; d < Dd; ++d) { float v = emb[i * Dd + d]; s += v * v; }
    nrm[i] = sqrtf(s);
  }
}

// ----------------------------------------------------------------- topk ----
// One block per row i. cos[i][j] for all j in LDS, then 20 argmax rounds
// (strict >, ties -> lower index; matches jax.lax.top_k stability).
__global__ __launch_bounds__(256) void topk_kernel(const float* __restrict__ emb,
                                                   const float* __restrict__ nrm,
                                                   int* __restrict__ topk) {
  __shared__ float cosv[Nd];
  __shared__ float redv[256];
  __shared__ int   redi[256];
  const int i = blockIdx.x;
  const int t = threadIdx.x;
  const float ni = nrm[i];
  for (int j = t; j < Nd; j += 256) {
    float dot = 0.f;
    #pragma unroll
    for (int d = 0; d < Dd; ++d) dot += emb[i * Dd + d] * emb[j * Dd + d];
    cosv[j] = dot / (ni * nrm[j]);
  }
  __syncthreads();
  for (int round = 0; round < TOPK; ++round) {
    float v0 = cosv[t], v1 = cosv[t + 256];
    float bv; int bi;
    if (v1 > v0) { bv = v1; bi = t + 256; } else { bv = v0; bi = t; }
    redv[t] = bv; redi[t] = bi;
    __syncthreads();
    for (int s = 128; s > 0; s >>= 1) {
      if (t < s) {
        float ov = redv[t + s]; int oi = redi[t + s];
        if (ov > redv[t] || (ov == redv[t] && oi < redi[t])) {
          redv[t] = ov; redi[t] = oi;
        }
      }
      __syncthreads();
    }
    if (t == 0) { topk[i * TOPK + round] = redi[0]; cosv[redi[0]] = -INFINITY; }
    __syncthreads();
  }
}

// ----------------------------------------------------- z GEMM via WMMA -----
// z[m][n] = sum_k data[m][k] * fc_w[n][k] + fc_b[n],  m in [0,65536), n,k in [0,64)
// One wave per 16-row M tile, covering all four 16-col tiles (D=64).
__global__ __launch_bounds__(256) void zgemm_kernel(const float* __restrict__ data,
                                                    const float* __restrict__ fcw,
                                                    const float* __restrict__ fcb,
                                                    float* __restrict__ z) {
  __shared__ float lfc[64 * 66];  // stride 66 floats: 8B aligned, no bank conflicts

#if __has_builtin(__builtin_amdgcn_global_load_async_to_lds_b64)
  // Direct memory -> LDS DMA (no VGPR round trip), tracked by ASYNCcnt.
  for (int idx = threadIdx.x; idx < 2048; idx += 256) {
    int r = idx >> 5, cp = idx & 31;
    __builtin_amdgcn_global_load_async_to_lds_b64(
        (gv2i_p)(void*)&fcw[r * 64 + 2 * cp],
        (lv2i_p)(void*)&lfc[r * 66 + 2 * cp],
        /*offset=*/0, /*cpol=*/0);
  }
#if __has_builtin(__builtin_amdgcn_s_wait_asynccnt)
  __builtin_amdgcn_s_wait_asynccnt(0);
#else
  asm volatile("s_wait_asynccnt 0" ::: "memory");
#endif
#else
  for (int idx = threadIdx.x; idx < 2048; idx += 256) {
    int r = idx >> 5, cp = idx & 31;
    *(float2*)&lfc[r * 66 + 2 * cp] = *(const float2*)&fcw[r * 64 + 2 * cp];
  }
#endif
  __syncthreads();

  const int lane = threadIdx.x & 31;
  const int wid  = threadIdx.x >> 5;
  const int half = lane >> 4;      // K group: 0 -> K={0,1}, 1 -> K={2,3}
  const int l16  = lane & 15;
  const size_t rowbase = ((size_t)blockIdx.x * 8 + wid) * 16;

  const float* A = data + (rowbase + (size_t)l16) * 64 + 2 * half;

  v8f acc[4];
  #pragma unroll
  for (int t = 0; t < 4; ++t) acc[t] = (v8f){0.f,0.f,0.f,0.f,0.f,0.f,0.f,0.f};

  #pragma unroll
  for (int k0 = 0; k0 < 64; k0 += 4) {
    v2f a = *(const v2f*)(A + k0);
    #pragma unroll
    for (int t = 0; t < 4; ++t) {
      v2f b = *(const v2f*)&lfc[(16 * t + l16) * 66 + k0 + 2 * half];
      acc[t] = __builtin_amdgcn_wmma_f32_16x16x4_f32(
          /*neg_a=*/false, a, /*neg_b=*/false, b,
          /*c_mod=*/(short)0, acc[t], /*reuse_a=*/false, /*reuse_b=*/false);
    }
  }

  #pragma unroll
  for (int t = 0; t < 4; ++t) {
    const int col = 16 * t + l16;
    const float bias = fcb[col];
    #pragma unroll
    for (int v = 0; v < 8; ++v) {
      size_t m = rowbase + (size_t)v + 8 * half;
      z[m * 64 + col] = acc[t][v] + bias;
    }
  }
}

// --------------------------------------------------------- attn scores -----
// s_src = z . attn_w[0:64]   + emb . attn_w[64:128]
// s_dst = z . attn_w[128:192]+ emb . attn_w[192:256]
__global__ void attn_kernel(const float* __restrict__ z,
                            const float* __restrict__ emb,
                            const float* __restrict__ aw,
                            float* __restrict__ ssrc,
                            float* __restrict__ sdst) {
  int t = blockIdx.x * blockDim.x + threadIdx.x;
  if (t >= PAIRS) return;
  int n = t & (Nd - 1);
  const float* zr = z + (size_t)t * Dd;
  const float* er = emb + (size_t)n * Dd;
  float as = 0.f, ad = 0.f;
  #pragma unroll
  for (int d = 0; d < Dd; ++d) {
    float zv = zr[d];
    as += zv * aw[d];
    ad += zv * aw[128 + d];
  }
  #pragma unroll
  for (int d = 0; d < Dd; ++d) {
    float ev = er[d];
    as += ev * aw[64 + d];
    ad += ev * aw[192 + d];
  }
  ssrc[t] = as;
  sdst[t] = ad;
}

// ------------------------------------------- softmax + aggregate + BN -----
// One wave per (b,n): lanes 0..19 hold the 20 incoming edges
// (edge k of dst n: src = topk_flat[n + 512*k]  -- reference's tile/reshape).
__global__ __launch_bounds__(256) void aggregate_kernel(
    const float* __restrict__ z, const float* __restrict__ emb,
    const float* __restrict__ ssrc, const float* __restrict__ sdst,
    const int* __restrict__ topk, const float* __restrict__ attn_b_p,
    float* __restrict__ rst, float* __restrict__ psum, float* __restrict__ psq) {
  const int lane = threadIdx.x & 31;
  const int wg = (blockIdx.x * blockDim.x + threadIdx.x) >> 5;
  const float ab = attn_b_p[0];
  float sum0 = 0.f, sum1 = 0.f, sq0 = 0.f, sq1 = 0.f;

  for (int it = 0; it < PAIRS_PER_WAVE; ++it) {
    const int pair = wg * PAIRS_PER_WAVE + it;
    const int b = pair >> 9;
    const int n = pair & (Nd - 1);

    int srcn = 0;
    float e = -INFINITY;
    if (lane < TOPK) {
      srcn = topk[n + (lane << 9)];
      float ev = ssrc[(b << 9) + srcn] + sdst[pair] + ab;
      e = (ev >= 0.f) ? ev : 0.2f * ev;   // leaky_relu
    }
    float m = e;
    #pragma unroll
    for (int o = 16; o; o >>= 1) m = fmaxf(m, __shfl_xor(m, o, 32));
    float p = (lane < TOPK) ? expf(e - m) : 0.f;
    float den = p;
    #pragma unroll
    for (int o = 16; o; o >>= 1) den += __shfl_xor(den, o, 32);
    const float alpha = p / den;

    float h0 = 0.f, h1 = 0.f;
    #pragma unroll
    for (int k = 0; k < TOPK; ++k) {
      float a = __shfl(alpha, k, 32);
      int s   = __shfl(srcn, k, 32);
      const float* zp = z + ((size_t)(b << 9) + (size_t)s) * Dd;
      h0 += a * zp[lane];
      h1 += a * zp[lane + 32];
    }
    const float r0 = h0 * emb[n * Dd + lane];
    const float r1 = h1 * emb[n * Dd + lane + 32];
    float* rp = rst + (size_t)pair * Dd;
    rp[lane] = r0;
    rp[lane + 32] = r1;
    sum0 += r0; sq0 += r0 * r0;
    sum1 += r1; sq1 += r1 * r1;
  }
  psum[wg * 64 + lane]      = sum0;
  psum[wg * 64 + lane + 32] = sum1;
  psq [wg * 64 + lane]      = sq0;
  psq [wg * 64 + lane + 32] = sq1;
}

// ------------------------------------------------------------- BN stats ----
__global__ void stats_kernel(const float* __restrict__ psum,
                             const float* __restrict__ psq,
                             const float* __restrict__ gamma,
                             const float* __restrict__ beta,
                             float* __restrict__ scale,
                             float* __restrict__ shift) {
  int d = threadIdx.x;
  if (d >= Dd) return;
  float s = 0.f, q = 0.f;
  for (int w = 0; w < AGG_WAVES; ++w) {  // fixed order -> deterministic
    s += psum[w * 64 + d];
    q += psq [w * 64 + d];
  }
  const float inv_n = 1.0f / (float)PAIRS;
  const float mu = s * inv_n;
  const float var = q * inv_n - mu * mu;
  const float sc = gamma[d] / sqrtf(var + 1e-5f);
  scale[d] = sc;
  shift[d] = beta[d] - mu * sc;
}

// ------------------------------------------------------------- output ------
__global__ void out_kernel(const float* __restrict__ rst,
                           const float* __restrict__ scale,
                           const float* __restrict__ shift,
                           const float* __restrict__ ow,
                           const float* __restrict__ ob_p,
                           float* __restrict__ out) {
  int t = blockIdx.x * blockDim.x + threadIdx.x;
  if (t >= PAIRS) return;
  const float* rp = rst + (size_t)t * Dd;
  float acc = ob_p[0];
  #pragma unroll
  for (int d = 0; d < Dd; ++d) {
    float r = rp[d] * scale[d] + shift[d];
    r = r > 0.f ? r : 0.f;
    acc += r * ow[d];
  }
  out[t] = acc;
}

// ---------------------------------------------------------------------------
extern "C" void kernel_launch(void* const* d_in, const int* in_sizes, int n_in,
                              void* d_out, int out_size, void* d_ws, size_t ws_size,
                              hipStream_t stream) {
  (void)in_sizes; (void)n_in; (void)out_size; (void)ws_size;
  const float* data   = (const float*)d_in[0];
  const float* emb    = (const float*)d_in[1];
  const float* fc_w   = (const float*)d_in[2];
  const float* fc_b   = (const float*)d_in[3];
  const float* attn_w = (const float*)d_in[4];
  const float* attn_b = (const float*)d_in[5];
  const float* gamma  = (const float*)d_in[6];
  const float* beta   = (const float*)d_in[7];
  const float* out_w  = (const float*)d_in[8];
  const float* out_b  = (const float*)d_in[9];
  float* out = (float*)d_out;

  // workspace layout (floats); total ~8.8M floats (~35.2 MB)
  float* ws    = (float*)d_ws;
  float* z     = ws;                         // 65536*64
  float* rst   = z + (size_t)PAIRS * Dd;     // 65536*64
  float* s_src = rst + (size_t)PAIRS * Dd;   // 65536
  float* s_dst = s_src + PAIRS;              // 65536
  float* nrm   = s_dst + PAIRS;              // 512
  int*   topk  = (int*)(nrm + Nd);           // 512*20
  float* psum  = (float*)(topk + Nd * TOPK); // 2048*64
  float* psq   = psum + AGG_WAVES * 64;      // 2048*64
  float* scale = psq + AGG_WAVES * 64;       // 64
  float* shift = scale + Dd;                 // 64

  norms_kernel<<<2, 256, 0, stream>>>(emb, nrm);
  topk_kernel<<<Nd, 256, 0, stream>>>(emb, nrm, topk);
  zgemm_kernel<<<512, 256, 0, stream>>>(data, fc_w, fc_b, z);
  attn_kernel<<<256, 256, 0, stream>>>(z, emb, attn_w, s_src, s_dst);
  aggregate_kernel<<<256, 256, 0, stream>>>(z, emb, s_src, s_dst, topk,
                                            attn_b, rst, psum, psq);
  stats_kernel<<<1, 64, 0, stream>>>(psum, psq, gamma, beta, scale, shift);
  out_kernel<<<256, 256, 0, stream>>>(rst, scale, shift, out_w, out_b, out);
}